// TtMambaSSM_73057393705596
// MI455X (gfx1250) — compile-verified
//
#include <hip/hip_runtime.h>
#include <hip/hip_bf16.h>
#include <math.h>

typedef __attribute__((ext_vector_type(2))) float v2f;
typedef __attribute__((ext_vector_type(4))) float v4f;
typedef __attribute__((ext_vector_type(8))) float v8f;

#define BATCH   512
#define DINNER  5120
#define DSTATE  16
#define DTRANK  160

// ---------------------------------------------------------------------------
// Generic fp32 WMMA GEMM: out[M,N] = A[M,K] * W[N,K]^T
// One 16x16 output tile per wave32, 8 waves per block.
// V_WMMA_F32_16X16X4_F32 fragment layout (ISA 7.12.2):
//   A 16x4:  lane m (0..15) holds K=k,k+1 in v0,v1; lane 16+m holds K=k+2,k+3
//   B 4x16:  lane n (0..15) holds K=k,k+1;          lane 16+n holds K=k+2,k+3
//   C/D:     vgpr i -> row m0+i (lanes 0..15) / m0+8+i (lanes 16..31), col = lane&15
// ---------------------------------------------------------------------------
__global__ __launch_bounds__(256) void gemm_xwt_f32(const float* __restrict__ A,
                                                    const float* __restrict__ W,
                                                    float* __restrict__ out,
                                                    int M, int N, int K) {
    const int lane = threadIdx.x & 31;
    const int wave = threadIdx.x >> 5;
    const int tile = blockIdx.x * 8 + wave;
    const int ntn  = N >> 4;
    if (tile >= (M >> 4) * ntn) return;           // wave-uniform, EXEC stays all-1s
    const int m0 = (tile / ntn) << 4;
    const int n0 = (tile % ntn) << 4;
    const int r  = lane & 15;
    const int kh = (lane >> 4) << 1;              // 0 or 2: which K-half this lane holds

    const float* Ap = A + (size_t)(m0 + r) * K + kh;
    const float* Wp = W + (size_t)(n0 + r) * K + kh;

    v8f acc = {0.f, 0.f, 0.f, 0.f, 0.f, 0.f, 0.f, 0.f};
#pragma unroll 4
    for (int k = 0; k < K; k += 4) {
        v2f av = *(const v2f*)(Ap + k);
        v2f bv = *(const v2f*)(Wp + k);
        acc = __builtin_amdgcn_wmma_f32_16x16x4_f32(false, av, false, bv,
                                                    (short)0, acc, false, false);
    }

    const int col  = n0 + r;
    const int row0 = m0 + ((lane >> 4) << 3);
#pragma unroll
    for (int i = 0; i < 8; ++i)
        out[(size_t)(row0 + i) * N + col] = acc[i];
}

// ---------------------------------------------------------------------------
// dt = softplus(dt_low @ W_dt^T + b_dt)   (same tiling, fused epilogue)
// ---------------------------------------------------------------------------
__global__ __launch_bounds__(256) void gemm_dt_softplus(const float* __restrict__ A,   // dt_low [512,160]
                                                        const float* __restrict__ W,   // W_dt  [5120,160]
                                                        const float* __restrict__ bias,// b_dt  [5120]
                                                        float* __restrict__ out,       // dt    [512,5120]
                                                        int M, int N, int K) {
    const int lane = threadIdx.x & 31;
    const int wave = threadIdx.x >> 5;
    const int tile = blockIdx.x * 8 + wave;
    const int ntn  = N >> 4;
    if (tile >= (M >> 4) * ntn) return;
    const int m0 = (tile / ntn) << 4;
    const int n0 = (tile % ntn) << 4;
    const int r  = lane & 15;
    const int kh = (lane >> 4) << 1;

    const float* Ap = A + (size_t)(m0 + r) * K + kh;
    const float* Wp = W + (size_t)(n0 + r) * K + kh;

    v8f acc = {0.f, 0.f, 0.f, 0.f, 0.f, 0.f, 0.f, 0.f};
#pragma unroll 4
    for (int k = 0; k < K; k += 4) {
        v2f av = *(const v2f*)(Ap + k);
        v2f bv = *(const v2f*)(Wp + k);
        acc = __builtin_amdgcn_wmma_f32_16x16x4_f32(false, av, false, bv,
                                                    (short)0, acc, false, false);
    }

    const int col  = n0 + r;
    const int row0 = m0 + ((lane >> 4) << 3);
    const float b  = bias[col];
#pragma unroll
    for (int i = 0; i < 8; ++i) {
        float v  = acc[i] + b;
        float sp = (v > 20.f) ? v : log1pf(__expf(v));   // softplus
        out[(size_t)(row0 + i) * N + col] = sp;
    }
}

// ---------------------------------------------------------------------------
// Streaming SSM update: one thread per (b,d).
//   y[b,d] = sum_s ( exp(A[d,s]*dt[b,d]) * h[b,d,s] + dt[b,d]*Bc[b,s]*x[b,d] ) * Cc[b,s]
//            + D[d]*x[b,d]
// h is stream-once (168 MB): nontemporal loads so x/dt/weights stay cached.
// ---------------------------------------------------------------------------
__global__ __launch_bounds__(256) void ssm_update(const float* __restrict__ xf,
                                                  const float* __restrict__ dt,
                                                  const float* __restrict__ Bc,
                                                  const float* __restrict__ Cc,
                                                  const float* __restrict__ Amat,
                                                  const float* __restrict__ Dvec,
                                                  const float* __restrict__ h,
                                                  float* __restrict__ y) {
    const int idx = blockIdx.x * 256 + threadIdx.x;     // flat (b, d)
    const int b   = idx / DINNER;
    const int d   = idx - b * DINNER;

    const float x   = xf[idx];
    const float dtv = dt[idx];
    const float dbx = dtv * x;

    const v4f* hp = (const v4f*)(h    + (size_t)idx * DSTATE);
    const v4f* Ap = (const v4f*)(Amat + (size_t)d   * DSTATE);
    const v4f* Bp = (const v4f*)(Bc   + (size_t)b   * DSTATE);
    const v4f* Cp = (const v4f*)(Cc   + (size_t)b   * DSTATE);

    float acc = 0.f;
#pragma unroll
    for (int j = 0; j < 4; ++j) {
        v4f hv = __builtin_nontemporal_load(hp + j);
        v4f av = Ap[j];
        v4f bv = Bp[j];
        v4f cv = Cp[j];
        acc += (__expf(av.x * dtv) * hv.x + dbx * bv.x) * cv.x;
        acc += (__expf(av.y * dtv) * hv.y + dbx * bv.y) * cv.y;
        acc += (__expf(av.z * dtv) * hv.z + dbx * bv.z) * cv.z;
        acc += (__expf(av.w * dtv) * hv.w + dbx * bv.w) * cv.w;
    }
    float r = acc + Dvec[d] * x;
    __builtin_nontemporal_store(r, y + idx);
}

// ---------------------------------------------------------------------------
extern "C" void kernel_launch(void* const* d_in, const int* in_sizes, int n_in,
                              void* d_out, int out_size, void* d_ws, size_t ws_size,
                              hipStream_t stream) {
    const float* x         = (const float*)d_in[0];   // [512,1,1,5120] == [512,5120]
    const float* W_dt_rank = (const float*)d_in[1];   // [160,5120]
    const float* W_dt      = (const float*)d_in[2];   // [5120,160]
    const float* b_dt      = (const float*)d_in[3];   // [5120]
    const float* W_B       = (const float*)d_in[4];   // [16,5120]
    const float* W_C       = (const float*)d_in[5];   // [16,5120]
    const float* A         = (const float*)d_in[6];   // [5120,16]
    const float* D         = (const float*)d_in[7];   // [5120]
    const float* h         = (const float*)d_in[8];   // [512,5120,16]
    float*       y         = (float*)d_out;           // [512,5120]

    char* ws = (char*)d_ws;
    float* dt_low = (float*)(ws);                                     // 512*160
    float* Bc     = (float*)(ws + ((size_t)BATCH * DTRANK) * 4);      // 512*16
    float* Cc     = Bc + (size_t)BATCH * DSTATE;
    float* dt     = Cc + (size_t)BATCH * DSTATE;                      // 512*5120

    // 1) dt_low = x @ W_dt_rank^T   (M=512, N=160, K=5120) -> 320 tiles
    gemm_xwt_f32<<<320 / 8, 256, 0, stream>>>(x, W_dt_rank, dt_low, BATCH, DTRANK, DINNER);
    // 2) Bc = x @ W_B^T, Cc = x @ W_C^T  (M=512, N=16, K=5120) -> 32 tiles each
    gemm_xwt_f32<<<32 / 8, 256, 0, stream>>>(x, W_B, Bc, BATCH, DSTATE, DINNER);
    gemm_xwt_f32<<<32 / 8, 256, 0, stream>>>(x, W_C, Cc, BATCH, DSTATE, DINNER);
    // 3) dt = softplus(dt_low @ W_dt^T + b_dt)  (M=512, N=5120, K=160) -> 10240 tiles
    gemm_dt_softplus<<<10240 / 8, 256, 0, stream>>>(dt_low, W_dt, b_dt, dt, BATCH, DINNER, DTRANK);
    // 4) streaming state update + output
    ssm_update<<<(BATCH * DINNER) / 256, 256, 0, stream>>>(x, dt, Bc, Cc, A, D, h, y);
}